// MultiExpertLoRALinearLayer_89816356094320
// MI455X (gfx1250) — compile-verified
//
#include <hip/hip_runtime.h>
#include <hip/hip_bf16.h>

// CDNA5 / gfx1250. Fused multi-expert LoRA:
//   C1[16384,64] = X[16384,1024] * DwT[1024,64]          (stacked expert down-proj)
//   C1 *= scale*alpha[b, rank_block]                      (folded per 16-col block)
//   Out[16384,1024] = C1 * U[64,1024]                     (stacked expert up-proj)
// All math in fp32 via V_WMMA_F32_16X16X4_F32 (exact fp32, f32 accumulate).
// Roofline: 4.3 GFLOP, 128 MB HBM traffic -> 5.5us floor; f32 WMMA ~300 TFLOPS -> ~14us.

typedef __attribute__((ext_vector_type(2))) float v2f;
typedef __attribute__((ext_vector_type(8))) float v8f;

#define IN_F   1024
#define OUT_F  1024
#define RANKS  64          // 4 experts * rank 16
#define ROWS   16384       // 4 * 4096
#define WAVES  4           // waves per workgroup
#define LDS_STRIDE 68      // 16-row tile row stride (floats); conflict-free b64 reads

// ---------------------------------------------------------------------------
// Pack weights into WMMA B-fragment layout (assumed, mirroring ISA A-layout):
//   v0: lanes0-15 -> row K=0 (N=lane), lanes16-31 -> row K=2 (N=lane-16)
//   v1: lanes0-15 -> row K=1,          lanes16-31 -> row K=3
// dwPack[dtile(256)][ntile(4)][lane(32)][j(2)]  from down_w[K=4,r=16,D=1024]
// upPack[rtile(16)][otile(64)][lane(32)][j(2)]  from up_w[K=4,out=1024,r=16]
// ---------------------------------------------------------------------------
__global__ void lora_pack_kernel(const float* __restrict__ down,
                                 const float* __restrict__ up,
                                 float* __restrict__ dwPack,
                                 float* __restrict__ upPack) {
  const int idx = blockIdx.x * blockDim.x + threadIdx.x;   // 0 .. 65535
  {
    const int j = idx & 1, lane = (idx >> 1) & 31;
    const int nt = (idx >> 6) & 3, dtile = idx >> 8;       // dtile < 256
    const int kk = ((lane >> 4) << 1) + j;                 // K-within-tile
    const int d = dtile * 4 + kk;
    const int n = nt * 16 + (lane & 15);                   // global rank 0..63
    // down_w[K,r,D] row-major: expert*16+rank == n
    dwPack[idx] = down[n * IN_F + d];
  }
  {
    const int j = idx & 1, lane = (idx >> 1) & 31;
    const int ot = (idx >> 6) & 63, rt = idx >> 12;        // rt < 16
    const int kk = ((lane >> 4) << 1) + j;
    const int r = rt * 4 + kk;                             // global rank 0..63
    const int o = ot * 16 + (lane & 15);
    // up_w[K,out,r] row-major
    upPack[idx] = up[((r >> 4) * OUT_F + o) * 16 + (r & 15)];
  }
}

// ---------------------------------------------------------------------------
// Fused GEMM1+scale+GEMM2. One wave owns 16 rows; 4 waves / 128-thread WG.
// LDS is wave-private -> no barriers (same-wave DS ops are in-order).
// ---------------------------------------------------------------------------
__global__ __launch_bounds__(WAVES * 32)
void lora_fused_kernel(const float* __restrict__ X,
                       const float* __restrict__ alpha,
                       const float* __restrict__ dwPack,
                       const float* __restrict__ upPack,
                       float* __restrict__ out) {
  __shared__ __align__(16) float ldsX[WAVES][16 * LDS_STRIDE];
  __shared__ __align__(16) float ldsC[WAVES][16 * LDS_STRIDE];

  const int lane = threadIdx.x & 31;
  const int w    = threadIdx.x >> 5;
  const int m0   = (blockIdx.x * WAVES + w) * 16;    // first global row of tile
  const int b    = m0 >> 12;                         // batch = m0 / 4096
  const int row  = lane & 15;                        // A-fragment row
  const int koff = (lane >> 4) << 1;                 // A-fragment K offset (0 or 2)

  v8f acc[4] = {v8f{}, v8f{}, v8f{}, v8f{}};         // C1 tile: 16 x 64

  // ---------------- GEMM1: C1 = X * DwT, K = 1024 in chunks of 64 ----------
  for (int c = 0; c < 16; ++c) {
    // Stage X[m0:m0+16, c*64 : c*64+64] into wave-private LDS (coalesced b128).
#pragma unroll
    for (int it = 0; it < 8; ++it) {
      const int idx = it * 32 + lane;
      const int r_ = idx >> 4, c4 = idx & 15;
      const float4 v = *reinterpret_cast<const float4*>(
          &X[(m0 + r_) * IN_F + c * 64 + c4 * 4]);
      *reinterpret_cast<float4*>(&ldsX[w][r_ * LDS_STRIDE + c4 * 4]) = v;
    }
    if (c < 15)  // global_prefetch_b8 of next chunk
      __builtin_prefetch(&X[(m0 + (lane >> 1)) * IN_F + (c + 1) * 64 + (lane & 1) * 32], 0, 0);

#pragma unroll
    for (int dt = 0; dt < 16; ++dt) {
      const v2f a = *reinterpret_cast<const v2f*>(
          &ldsX[w][row * LDS_STRIDE + dt * 4 + koff]);
#pragma unroll
      for (int nt = 0; nt < 4; ++nt) {
        const v2f bf = *reinterpret_cast<const v2f*>(
            &dwPack[(((c * 16 + dt) * 4 + nt) * 32 + lane) * 2]);
        acc[nt] = __builtin_amdgcn_wmma_f32_16x16x4_f32(
            false, a, false, bf, (short)0, acc[nt], false, false);
      }
    }
  }

  // -------- scale by network_alpha/rank * alpha[b,expert]; C1 -> LDS -------
#pragma unroll
  for (int nt = 0; nt < 4; ++nt) {
    const float s = 4.0f * alpha[b * 4 + nt];        // 64.0/16 = 4.0, expert==nt
#pragma unroll
    for (int j = 0; j < 8; ++j) {
      const int mrow = (lane < 16) ? j : (j + 8);    // C/D layout (ISA 7.12.2)
      ldsC[w][mrow * LDS_STRIDE + nt * 16 + (lane & 15)] = acc[nt][j] * s;
    }
  }

  // Preload all 16 A-fragments of C1 (K = 64 ranks, 16 VGPR pairs).
  v2f a2[16];
#pragma unroll
  for (int rt = 0; rt < 16; ++rt)
    a2[rt] = *reinterpret_cast<const v2f*>(
        &ldsC[w][row * LDS_STRIDE + rt * 4 + koff]);

  // ---------------- GEMM2: Out = C1 * U, stream 64 output tiles ------------
  for (int ot = 0; ot < 64; ++ot) {
    v8f accO = v8f{};
#pragma unroll
    for (int rt = 0; rt < 16; ++rt) {
      const v2f bf = *reinterpret_cast<const v2f*>(
          &upPack[((rt * 64 + ot) * 32 + lane) * 2]);
      accO = __builtin_amdgcn_wmma_f32_16x16x4_f32(
          false, a2[rt], false, bf, (short)0, accO, false, false);
    }
#pragma unroll
    for (int j = 0; j < 8; ++j) {
      const int mrow = (lane < 16) ? j : (j + 8);
      out[(m0 + mrow) * OUT_F + ot * 16 + (lane & 15)] = accO[j];
    }
  }
}

extern "C" void kernel_launch(void* const* d_in, const int* in_sizes, int n_in,
                              void* d_out, int out_size, void* d_ws, size_t ws_size,
                              hipStream_t stream) {
  const float* hs    = (const float*)d_in[0];   // [4,4096,1024]
  const float* alpha = (const float*)d_in[1];   // [4,4]
  const float* down  = (const float*)d_in[2];   // [4,16,1024]
  const float* up    = (const float*)d_in[3];   // [4,1024,16]
  float* out = (float*)d_out;                   // [4,4096,1024]

  float* dwPack = (float*)d_ws;                 // 65536 floats = 256 KB
  float* upPack = dwPack + 65536;               // 65536 floats = 256 KB

  lora_pack_kernel<<<256, 256, 0, stream>>>(down, up, dwPack, upPack);

  const int nblocks = ROWS / (WAVES * 16);      // 256 workgroups
  lora_fused_kernel<<<nblocks, WAVES * 32, 0, stream>>>(hs, alpha, dwPack, upPack, out);
}